// PrPool_60266981097823
// MI455X (gfx1250) — compile-verified
//
#include <hip/hip_runtime.h>
#include <cstdint>
#include <cstddef>

typedef __attribute__((ext_vector_type(2))) float v2f;
typedef __attribute__((ext_vector_type(8))) float v8f;

namespace {
constexpr int   kC     = 256;
constexpr int   kHf    = 56;
constexpr int   kWf    = 56;
constexpr int   kWxP   = 60;          // wx row stride, padded with zeros (cols 56..59)
constexpr int   kPh    = 7;
constexpr int   kPw    = 7;
constexpr int   kPQ    = kPh * kPw;   // 49
constexpr float kScale = 0.0625f;     // 1/16 backbone stride
}

// Antiderivative of the bilinear hat kernel on [-1,1]
__device__ __forceinline__ float hat_H(float s) {
  return (s < 0.0f) ? 0.5f * (s + 1.0f) * (s + 1.0f)
                    : 1.0f - 0.5f * (1.0f - s) * (1.0f - s);
}
// Integral of max(0, 1-|t-n|) over [a,b] (with clipping, as in reference)
__device__ __forceinline__ float hat_int(float a, float b, float n) {
  float sa = fminf(fmaxf(a - n, -1.0f), 1.0f);
  float sb = fminf(fmaxf(b - n, -1.0f), 1.0f);
  return hat_H(sb) - hat_H(sa);
}

// One workgroup per ROI. 256 threads = 8 waves (wave32).
// GEMM: out[256 x 49] = F[256 x K] * W[K x 49], K = clipped window(i,j), via
// v_wmma_f32_16x16x4_f32. Wave w owns C-tiles {w, w+8}; each C-tile keeps
// 4 N-tile accumulators (N=49 padded to 64) and reuses A across them.
// wy table is pre-scaled by 1/area; wx table is zero-padded so padded K
// columns contribute exactly zero without per-chunk masking.
__global__ __launch_bounds__(256, 1)
void prroi_pool_wmma(const float* __restrict__ input,
                     const float* __restrict__ rois,
                     float* __restrict__ out) {
  __shared__ float wy_s[kPh * kHf];    // [p][i], pre-scaled by oscale
  __shared__ float wx_s[kPw * kWxP];   // [q][j], j in [0,60), zeros for j>=56

  const int   r  = blockIdx.x;
  const float bf = rois[r * 5 + 0];
  const float x1 = rois[r * 5 + 1] * kScale;
  const float y1 = rois[r * 5 + 2] * kScale;
  const float x2 = rois[r * 5 + 3] * kScale;
  const float y2 = rois[r * 5 + 4] * kScale;

  const float bin_w  = fmaxf(x2 - x1, 0.0f) * (1.0f / kPw);
  const float bin_h  = fmaxf(y2 - y1, 0.0f) * (1.0f / kPh);
  const float area   = bin_w * bin_h;
  const float oscale = (area > 0.0f) ? 1.0f / fmaxf(area, 1e-12f) : 0.0f;

  // ---- Stage 1: weight tables into LDS (392 + 420 entries) ----
  for (int t = (int)threadIdx.x; t < kPh * kHf + kPw * kWxP; t += (int)blockDim.x) {
    if (t < kPh * kHf) {
      const int p = t / kHf, i = t % kHf;
      const float a = y1 + (float)p * bin_h;
      wy_s[t] = hat_int(a, a + bin_h, (float)i) * oscale;   // fold 1/area here
    } else {
      const int tt = t - kPh * kHf;
      const int q = tt / kWxP, j = tt % kWxP;
      const float a = x1 + (float)q * bin_w;
      wx_s[tt] = (j < kWf) ? hat_int(a, a + bin_w, (float)j) : 0.0f;
    }
  }
  __syncthreads();

  // Clipped window covering the whole hat-kernel support. These are uniform
  // per block (derived from the same ROI) — force them into SGPRs so both
  // GEMM loops use scalar control flow (no exec-mask loop handling).
  const int i0 = __builtin_amdgcn_readfirstlane(max(0, (int)floorf(y1) - 1));
  const int i1 = __builtin_amdgcn_readfirstlane(min(kHf - 1, (int)ceilf(y2) + 1));
  const int j0 = __builtin_amdgcn_readfirstlane(max(0, (int)floorf(x1) - 1));
  const int j1 = __builtin_amdgcn_readfirstlane(min(kWf - 1, (int)ceilf(x2) + 1));
  const int bi = __builtin_amdgcn_readfirstlane((int)bf);

  const int lane = (int)(threadIdx.x & 31u);
  const int wid  = (int)(threadIdx.x >> 5);
  const int m    = lane & 15;            // M index (A) / N index (B,D)
  const int khi  = (lane >> 4) << 1;     // lanes 0-15 -> K {0,1}; 16-31 -> K {2,3}
  const int rofs = (lane >> 4) << 3;     // D rows: v (lanes<16) / v+8 (lanes>=16)

  // Per-lane (p,q) row bases for the 4 N-tiles; padded n>=49 masked to zero.
  int   wyb[4], wxb[4];
  float nmask[4];
#pragma unroll
  for (int nt = 0; nt < 4; ++nt) {
    const int  n = nt * 16 + m;
    const bool v = (n < kPQ);
    const int  p = v ? (n / kPw) : 0;
    const int  q = v ? (n % kPw) : 0;
    wyb[nt]   = p * kHf;
    wxb[nt]   = q * kWxP;
    nmask[nt] = v ? 1.0f : 0.0f;
  }

  const float* feat = input + (size_t)bi * kC * kHf * kWf;

  for (int ct = 0; ct < 2; ++ct) {
    const int c0 = (wid + ct * 8) * 16;
    const float* fch = feat + (size_t)(c0 + m) * (kHf * kWf);

    v8f acc0 = {}, acc1 = {}, acc2 = {}, acc3 = {};

    for (int i = i0; i <= i1; ++i) {
      const float* frow = fch + i * kWf;
      const float wy0 = nmask[0] * wy_s[wyb[0] + i];
      const float wy1 = nmask[1] * wy_s[wyb[1] + i];
      const float wy2 = nmask[2] * wy_s[wyb[2] + i];
      const float wy3 = nmask[3] * wy_s[wyb[3] + i];

      for (int jc = j0; jc <= j1; jc += 4) {
        const int jA  = jc + khi;          // this lane's two K columns
        const int jB  = jA + 1;
        const int jAc = min(jA, kWf - 1);  // safe address; value don't-care when
        const int jBc = min(jB, kWf - 1);  // padded (wx reads exact 0 there)

        // A: 16x4 f32 tile (per ISA layout): lane m, K pair selected by khi
        v2f a;
        a.x = frow[jAc];
        a.y = frow[jBc];

        // B: 4x16 f32 tile, built on the fly: B[k, pq] = wy[p,i] * wx[q,j(k)]
        // (wx rows zero-padded to j<60, so jA/jB up to j1+3 are in-bounds)
        v2f b0, b1, b2, b3;
        b0.x = wy0 * wx_s[wxb[0] + jA];  b0.y = wy0 * wx_s[wxb[0] + jB];
        b1.x = wy1 * wx_s[wxb[1] + jA];  b1.y = wy1 * wx_s[wxb[1] + jB];
        b2.x = wy2 * wx_s[wxb[2] + jA];  b2.y = wy2 * wx_s[wxb[2] + jB];
        b3.x = wy3 * wx_s[wxb[3] + jA];  b3.y = wy3 * wx_s[wxb[3] + jB];

        acc0 = __builtin_amdgcn_wmma_f32_16x16x4_f32(false, a, false, b0,
                                                     (short)0, acc0, false, false);
        acc1 = __builtin_amdgcn_wmma_f32_16x16x4_f32(false, a, false, b1,
                                                     (short)0, acc1, false, false);
        acc2 = __builtin_amdgcn_wmma_f32_16x16x4_f32(false, a, false, b2,
                                                     (short)0, acc2, false, false);
        acc3 = __builtin_amdgcn_wmma_f32_16x16x4_f32(false, a, false, b3,
                                                     (short)0, acc3, false, false);
      }
    }

    // ---- Epilogue: D layout = VGPR v -> row c0+rofs+v, col n = nt*16+m ----
    // 1/area already folded into wy table; store raw accumulators.
#define PRROI_STORE_TILE(ACC, NT)                                            \
    do {                                                                     \
      const int n = (NT) * 16 + m;                                           \
      if (n < kPQ) {                                                         \
        float* o = out + ((size_t)r * kC + (size_t)(c0 + rofs)) * kPQ + n;   \
        _Pragma("unroll")                                                    \
        for (int v = 0; v < 8; ++v) o[(size_t)v * kPQ] = (ACC)[v];           \
      }                                                                      \
    } while (0)

    PRROI_STORE_TILE(acc0, 0);
    PRROI_STORE_TILE(acc1, 1);
    PRROI_STORE_TILE(acc2, 2);
    PRROI_STORE_TILE(acc3, 3);
#undef PRROI_STORE_TILE
  }
}

extern "C" void kernel_launch(void* const* d_in, const int* in_sizes, int n_in,
                              void* d_out, int out_size, void* d_ws, size_t ws_size,
                              hipStream_t stream) {
  const float* input = (const float*)d_in[0];   // [B, C, Hf, Wf] f32
  const float* rois  = (const float*)d_in[1];   // [R, 5] f32
  float* out = (float*)d_out;                   // [R, C, 7, 7] f32

  const int R = in_sizes[1] / 5;                // 128 ROIs
  prroi_pool_wmma<<<dim3((unsigned)R), dim3(256), 0, stream>>>(input, rois, out);

  (void)n_in; (void)out_size; (void)d_ws; (void)ws_size;
}